// DWT_34789235097770
// MI455X (gfx1250) — compile-verified
//
#include <hip/hip_runtime.h>
#include <stdint.h>

// Problem geometry (fixed by the reference).
#define BC        48          // 16 * 3 images
#define W_IN      1024
#define H_IN      1024
#define W_OUT     512
#define H_OUT     512
#define ROWS_PB   8           // output rows per block
#define THREADS   512         // one thread per output column
#define CHUNKS    (H_OUT / ROWS_PB)   // 64

typedef unsigned int u32x4 __attribute__((ext_vector_type(4)));
typedef int          i32x4 __attribute__((ext_vector_type(4)));
typedef int          i32x8 __attribute__((ext_vector_type(8)));

#if __has_include(<hip/amd_detail/amd_gfx1250_TDM.h>)
#define TDM_6ARG 1
#endif

#if defined(__gfx1250__) && __has_builtin(__builtin_amdgcn_tensor_load_to_lds)
#define USE_TDM 1
#endif

#if defined(__gfx1250__) && __has_builtin(__builtin_amdgcn_s_wait_tensorcnt)
#define WAIT_TENSORCNT(n) __builtin_amdgcn_s_wait_tensorcnt(n)
#elif defined(__gfx1250__)
#define WAIT_TENSORCNT(n) asm volatile("s_wait_tensorcnt %0" ::"i"(n))
#else
#define WAIT_TENSORCNT(n)
#endif

// Non-temporal store: outputs are written once and never re-read; keep them
// from rinsing through the 192MB L2 so the TDM read stream keeps its capacity.
#if __has_builtin(__builtin_nontemporal_store)
#define NT_STORE(v, p) __builtin_nontemporal_store((v), (p))
#else
#define NT_STORE(v, p) (*(p) = (v))
#endif

#ifdef USE_TDM
// Issue one TDM load of a (nrows x 1024) f32 tile (row stride 1024 elements)
// from global address gsrc into LDS byte offset lds_byte. Wave-level op
// (EXEC ignored) -- call from exactly one wave of the block.
__device__ __forceinline__ void tdm_load_rows(const float* gsrc,
                                              uint32_t lds_byte, int nrows) {
  uint64_t ga = (uint64_t)(uintptr_t)gsrc;
  u32x4 g0;
  g0[0] = 1u;                                   // count=1 (valid), user mode
  g0[1] = lds_byte;                             // lds_addr [63:32]
  g0[2] = (uint32_t)(ga & 0xFFFFFFFFu);         // global_addr lo
  g0[3] = ((uint32_t)(ga >> 32) & 0x01FFFFFFu)  // global_addr[56:32]
          | (2u << 30);                         // type=2 ("image")
  i32x8 g1;
  g1[0] = (int)(2u << 16);                      // data_size=2 -> 4 bytes
  g1[1] = (int)((W_IN & 0xFFFF) << 16);         // tensor_dim0 lo16 @ [63:48]
  g1[2] = (int)(((W_IN >> 16) & 0xFFFF)         // tensor_dim0 hi16
          | ((nrows & 0xFFFF) << 16));          // tensor_dim1 lo16
  g1[3] = (int)(((unsigned)nrows >> 16)         // tensor_dim1 hi16
          | ((W_IN & 0xFFFF) << 16));           // tile_dim0 = 1024
  g1[4] = nrows;                                // tile_dim1 = nrows, tile_dim2=0
  g1[5] = W_IN;                                 // tensor_dim0_stride lo32 (elems)
  g1[6] = 0;                                    // stride hi / dim1_stride lo
  g1[7] = 0;                                    // dim1_stride hi
  i32x4 z4 = {0, 0, 0, 0};                      // groups 2/3 unused (2D tensor)
#ifdef TDM_6ARG
  i32x8 z8 = {0, 0, 0, 0, 0, 0, 0, 0};
  __builtin_amdgcn_tensor_load_to_lds(g0, g1, z4, z4, z8, 0);
#else
  __builtin_amdgcn_tensor_load_to_lds(g0, g1, z4, z4, 0);
#endif
}
#endif

__global__ __launch_bounds__(THREADS)
void dwt_haar_kernel(const float* __restrict__ x, float* __restrict__ out) {
  __shared__ float buf[2][2 * W_IN];   // double-buffered 2-row stages (16 KB)

  const int   bc = blockIdx.x / CHUNKS;
  const int   i0 = (blockIdx.x % CHUNKS) * ROWS_PB;
  const int   j  = threadIdx.x;                  // output column 0..511
  const float S  = 0.70710678118654752440f;      // 1/sqrt(2), matches reference
  const size_t plane = (size_t)BC * H_OUT * W_OUT;

  const float* img = x + (size_t)bc * H_IN * W_IN;
  float* ll = out;
  float* lh = out + plane;
  float* hl = out + 2 * plane;
  float* hh = out + 3 * plane;

#ifdef USE_TDM
  const bool issuer = (threadIdx.x < 32);        // wave 0 (wave32)
  const uint32_t lds0 = (uint32_t)(uintptr_t)&buf[0][0];
  const uint32_t lds1 = (uint32_t)(uintptr_t)&buf[1][0];

  if (issuer) {                                  // prologue: stage 0
    const int i = i0;
    const int r0 = (i > 0) ? (2 * i - 1) : 0;
    tdm_load_rows(img + (size_t)r0 * W_IN, lds0, 2);
  }
  for (int s = 0; s < ROWS_PB; ++s) {
    const int i = i0 + s;
    if (issuer) {
      if (s + 1 < ROWS_PB) {                     // prefetch next row pair
        const int i2 = i0 + s + 1;               // i2 >= 1 always
        tdm_load_rows(img + (size_t)(2 * i2 - 1) * W_IN,
                      ((s + 1) & 1) ? lds1 : lds0, 2);
        WAIT_TENSORCNT(1);                       // stage s complete
      } else {
        WAIT_TENSORCNT(0);
      }
    }
    __syncthreads();                             // stage s visible to all waves

    const float* rb  = buf[s & 1];
    const bool  has_prev = (i > 0);
    const float* prv = rb;                       // row 2i-1 (valid iff has_prev)
    const float* cur = has_prev ? (rb + W_IN) : rb;  // row 2i

    float c = (j > 0) ? cur[2 * j - 1] : 0.f;
    float d = cur[2 * j];
    float a = 0.f, b = 0.f;
    if (has_prev) {
      b = prv[2 * j];
      if (j > 0) a = prv[2 * j - 1];
    }
    // Two-stage filtering, same operation order as the reference convs.
    const float lp = S * a + S * b;              // horiz lowpass, prev row
    const float hp = S * b - S * a;              // horiz highpass, prev row
    const float lc = S * c + S * d;              // horiz lowpass, cur row
    const float hc = S * d - S * c;              // horiz highpass, cur row
    const size_t o = ((size_t)bc * H_OUT + i) * W_OUT + j;
    NT_STORE(S * lp + S * lc, &ll[o]);
    NT_STORE(S * lc - S * lp, &lh[o]);
    NT_STORE(S * hp + S * hc, &hl[o]);
    NT_STORE(S * hc - S * hp, &hh[o]);
    __syncthreads();                             // buf[s&1] reusable next+1 iter
  }
#else
  // Fallback (non-gfx1250 pass / builtin missing): coalesced float4 staging.
  for (int s = 0; s < ROWS_PB; ++s) {
    const int i = i0 + s;
    const int r0 = (i > 0) ? (2 * i - 1) : 0;
    const float4* src = (const float4*)(img + (size_t)r0 * W_IN);
    ((float4*)buf[0])[threadIdx.x] = src[threadIdx.x];  // 2048 f = 512 float4
    __syncthreads();
    const bool has_prev = (i > 0);
    const float* prv = buf[0];
    const float* cur = has_prev ? (buf[0] + W_IN) : buf[0];
    float c = (j > 0) ? cur[2 * j - 1] : 0.f;
    float d = cur[2 * j];
    float a = 0.f, b = 0.f;
    if (has_prev) {
      b = prv[2 * j];
      if (j > 0) a = prv[2 * j - 1];
    }
    const float lp = S * a + S * b;
    const float hp = S * b - S * a;
    const float lc = S * c + S * d;
    const float hc = S * d - S * c;
    const size_t o = ((size_t)bc * H_OUT + i) * W_OUT + j;
    NT_STORE(S * lp + S * lc, &ll[o]);
    NT_STORE(S * lc - S * lp, &lh[o]);
    NT_STORE(S * hp + S * hc, &hl[o]);
    NT_STORE(S * hc - S * hp, &hh[o]);
    __syncthreads();
  }
#endif
}

extern "C" void kernel_launch(void* const* d_in, const int* in_sizes, int n_in,
                              void* d_out, int out_size, void* d_ws,
                              size_t ws_size, hipStream_t stream) {
  const float* x = (const float*)d_in[0];   // (16,3,1024,1024) f32
  float* out = (float*)d_out;               // LL|LH|HL|HH, each (16,3,512,512)
  (void)in_sizes; (void)n_in; (void)out_size; (void)d_ws; (void)ws_size;
  dim3 grid(BC * CHUNKS);                   // 3072 blocks
  dwt_haar_kernel<<<grid, THREADS, 0, stream>>>(x, out);
}